// TaxFraudGAT_33792802685031
// MI455X (gfx1250) — compile-verified
//
#include <hip/hip_runtime.h>
#include <hip/hip_bf16.h>
#include <math.h>

// ---------------- problem constants (match reference) ----------------
#define N_NODES   50000
#define N_EDGES   500000
#define EPRIME    (N_EDGES + N_NODES)   // edges + self loops = 550000
#define F_NODE    64
#define F_EDGE    16
#define HID       64
#define HEADS     4
#define NEG_SLOPE 0.2f

typedef __attribute__((ext_vector_type(2))) float v2f;
typedef __attribute__((ext_vector_type(8))) float v8f;

#define CEILDIV(a,b) (((a)+(b)-1)/(b))

// ---- order-preserving float->uint encoding for atomic max on scores ----
__device__ __forceinline__ unsigned fenc(float f) {
    unsigned u = __float_as_uint(f);
    return (u & 0x80000000u) ? ~u : (u | 0x80000000u);
}
__device__ __forceinline__ float fdec(unsigned u) {
    return (u & 0x80000000u) ? __uint_as_float(u ^ 0x80000000u)
                             : __uint_as_float(~u);
}

// ---------------------------------------------------------------------
__global__ void zero_f32(float* __restrict__ p, long long n) {
    long long i = (long long)blockIdx.x * blockDim.x + threadIdx.x;
    if (i < n) p[i] = 0.0f;
}

// ---------------------------------------------------------------------
// edge_attr mean: local accumulation + few atomics
__global__ void easum_kernel(const float* __restrict__ ea, float* __restrict__ easum) {
    float loc[F_EDGE];
#pragma unroll
    for (int f = 0; f < F_EDGE; ++f) loc[f] = 0.0f;
    for (long long e = (long long)blockIdx.x * blockDim.x + threadIdx.x;
         e < N_EDGES; e += (long long)gridDim.x * blockDim.x) {
        const float4* p = (const float4*)(ea + e * F_EDGE);
        float4 t0 = p[0], t1 = p[1], t2 = p[2], t3 = p[3];
        loc[0]+=t0.x; loc[1]+=t0.y; loc[2]+=t0.z; loc[3]+=t0.w;
        loc[4]+=t1.x; loc[5]+=t1.y; loc[6]+=t1.z; loc[7]+=t1.w;
        loc[8]+=t2.x; loc[9]+=t2.y; loc[10]+=t2.z; loc[11]+=t2.w;
        loc[12]+=t3.x; loc[13]+=t3.y; loc[14]+=t3.z; loc[15]+=t3.w;
    }
#pragma unroll
    for (int f = 0; f < F_EDGE; ++f) atomicAdd(&easum[f], loc[f]);
}
__global__ void eamean_div_kernel(float* __restrict__ easum) {
    if (threadIdx.x < F_EDGE) easum[threadIdx.x] *= (1.0f / (float)N_EDGES);
}

// ---------------------------------------------------------------------
// fp32 WMMA GEMM with LDS-staged weight panel.
// Block = 256 threads (8 waves). Block computes a 128-row x 64-col panel:
// each wave owns 16 rows x 64 cols = 4 accumulators of 16x16.
// B panel (K x 64) staged in LDS as float2 {B[k][c], B[k+1][c]} so the
// inner loop is 1 global_load_b64 (A pair) + 4 ds_load_b64 feeding 4 WMMAs.
// A 16x4 layout (ISA 7.12.2): lanes 0-15 hold {k0,k0+1} of row=lane,
//                             lanes 16-31 hold {k0+2,k0+3} of row=lane-16.
// C/D: VGPR v -> M = v (+8 for lanes 16-31), N = lane&15.
__global__ void gemm_wmma_lds(const float* __restrict__ A, const float* __restrict__ B,
                              const float* __restrict__ bias, float* __restrict__ C,
                              int M, int K, int Nout, int act) {
    extern __shared__ float2 Bs[];              // (K/2) * 64 float2
    const int lane = threadIdx.x & 31;
    const int wave = threadIdx.x >> 5;
    const int cgrp = Nout >> 6;                 // 64-col groups
    const int rg   = blockIdx.x / cgrp;
    const int cg   = blockIdx.x % cgrp;
    const int colbase = cg * 64;

    // cooperative stage of the weight panel (coalesced along columns)
    const int halfK = K >> 1;
    for (int idx = threadIdx.x; idx < halfK * 64; idx += blockDim.x) {
        int kk = idx >> 6, c = idx & 63;
        float2 t;
        t.x = B[(size_t)(2 * kk)     * Nout + colbase + c];
        t.y = B[(size_t)(2 * kk + 1) * Nout + colbase + c];
        Bs[idx] = t;
    }
    __syncthreads();

    const int row0 = rg * 128 + wave * 16;
    if (row0 >= M) return;                      // wave-uniform tail exit

    const int r  = lane & 15;
    const int kh = (lane >> 4) << 1;            // 0 or 2
    const float* Arow = A + (size_t)(row0 + r) * K;
    v8f acc[4] = {{}, {}, {}, {}};
    for (int k0 = 0; k0 < K; k0 += 4) {
        const int k = k0 + kh;                  // even
        v2f a;
        a.x = Arow[k];
        a.y = Arow[k + 1];
        const float2* bp = Bs + ((k >> 1) << 6);
#pragma unroll
        for (int t = 0; t < 4; ++t) {
            float2 bt = bp[t * 16 + r];
            v2f b; b.x = bt.x; b.y = bt.y;
            acc[t] = __builtin_amdgcn_wmma_f32_16x16x4_f32(false, a, false, b,
                                                           (short)0, acc[t], false, false);
        }
    }
    const int rbase = row0 + ((lane >> 4) << 3);
#pragma unroll
    for (int t = 0; t < 4; ++t) {
        const int col = colbase + t * 16 + r;
        const float bv = bias ? bias[col] : 0.0f;
#pragma unroll
        for (int v = 0; v < 8; ++v) {
            float val = acc[t][v] + bv;
            if (act == 1) val = fmaxf(val, 0.0f);
            C[(size_t)(rbase + v) * Nout + col] = val;
        }
    }
}

// ---------------------------------------------------------------------
// Layer-1 scores: wave per edge; lane handles 8 channels (1 head = 8 lanes).
// Fuses ee = ea @ We1 into the score; atomicMax into encoded smax.
__global__ void scores1_kernel(const float* __restrict__ xl, const float* __restrict__ xr,
                               const int* __restrict__ src, const int* __restrict__ dst,
                               const float* __restrict__ ea, const float* __restrict__ eamean,
                               const float* __restrict__ We1, const float* __restrict__ att1,
                               float* __restrict__ s1, unsigned* __restrict__ smax1) {
    int e = blockIdx.x * (blockDim.x >> 5) + (threadIdx.x >> 5);
    if (e >= EPRIME) return;
    const int lane = threadIdx.x & 31;
    int sn, dn; const float* eap;
    if (e < N_EDGES) { sn = src[e]; dn = dst[e]; eap = ea + (size_t)e * F_EDGE; }
    else             { sn = dn = e - N_EDGES;    eap = eamean; }
    float eav[F_EDGE];
#pragma unroll
    for (int f = 0; f < F_EDGE; ++f) eav[f] = eap[f];
    const int col0 = lane * 8;
    const float4* xl4 = (const float4*)(xl + (size_t)sn * 256 + col0);
    const float4* xr4 = (const float4*)(xr + (size_t)dn * 256 + col0);
    float4 a0 = xl4[0], a1 = xl4[1];
    float4 b0 = xr4[0], b1 = xr4[1];
    float mv[8] = { a0.x + b0.x, a0.y + b0.y, a0.z + b0.z, a0.w + b0.w,
                    a1.x + b1.x, a1.y + b1.y, a1.z + b1.z, a1.w + b1.w };
    float acc = 0.0f;
#pragma unroll
    for (int j = 0; j < 8; ++j) {
        int col = col0 + j;
        float eev = 0.0f;
#pragma unroll
        for (int f = 0; f < F_EDGE; ++f) eev += eav[f] * We1[f * 256 + col];
        float m = mv[j] + eev;
        m = (m > 0.0f) ? m : NEG_SLOPE * m;
        acc += m * att1[col];                 // att1 flat: h*64 + c == col
    }
    acc += __shfl_xor(acc, 1, 32);
    acc += __shfl_xor(acc, 2, 32);
    acc += __shfl_xor(acc, 4, 32);
    if ((lane & 7) == 0) {
        int h = lane >> 3;
        s1[(size_t)e * HEADS + h] = acc;
        atomicMax(&smax1[(size_t)dn * HEADS + h], fenc(acc));
    }
}

__global__ void expden1_kernel(const int* __restrict__ dst, float* __restrict__ s1,
                               const unsigned* __restrict__ smax1, float* __restrict__ den1) {
    long long i = (long long)blockIdx.x * blockDim.x + threadIdx.x;
    if (i >= (long long)EPRIME * HEADS) return;
    int e = (int)(i >> 2), h = (int)(i & 3);
    int dn = (e < N_EDGES) ? dst[e] : e - N_EDGES;
    float ez = __expf(s1[i] - fdec(smax1[(size_t)dn * HEADS + h]));
    s1[i] = ez;
    atomicAdd(&den1[(size_t)dn * HEADS + h], ez);
}

__global__ void scatter1_kernel(const float* __restrict__ xl,
                                const int* __restrict__ src, const int* __restrict__ dst,
                                const float* __restrict__ s1, const float* __restrict__ den1,
                                float* __restrict__ out1) {
    int e = blockIdx.x * (blockDim.x >> 5) + (threadIdx.x >> 5);
    if (e >= EPRIME) return;
    const int lane = threadIdx.x & 31;
    int sn, dn;
    if (e < N_EDGES) { sn = src[e]; dn = dst[e]; } else { sn = dn = e - N_EDGES; }
    int h = lane >> 3;
    float alpha = s1[(size_t)e * HEADS + h] / (den1[(size_t)dn * HEADS + h] + 1e-16f);
    const float* xs = xl + (size_t)sn * 256 + lane * 8;
    float* od = out1 + (size_t)dn * 256 + lane * 8;
#pragma unroll
    for (int j = 0; j < 8; ++j) atomicAdd(&od[j], alpha * xs[j]);
}

__global__ void addbias_elu_kernel(const float* __restrict__ in, const float* __restrict__ bias,
                                   float* __restrict__ out, long long n, int cmask) {
    long long i = (long long)blockIdx.x * blockDim.x + threadIdx.x;
    if (i >= n) return;
    float v = in[i] + bias[i & cmask];
    out[i] = (v > 0.0f) ? v : (__expf(v) - 1.0f);
}

// ---------------------------------------------------------------------
// Layer-2 (heads=1, C=64): wave per edge, 2 channels per lane.
__global__ void scores2_kernel(const float* __restrict__ xl, const float* __restrict__ xr,
                               const int* __restrict__ src, const int* __restrict__ dst,
                               const float* __restrict__ ea, const float* __restrict__ eamean,
                               const float* __restrict__ We2, const float* __restrict__ att2,
                               float* __restrict__ s2, unsigned* __restrict__ smax2) {
    int e = blockIdx.x * (blockDim.x >> 5) + (threadIdx.x >> 5);
    if (e >= EPRIME) return;
    const int lane = threadIdx.x & 31;
    int sn, dn; const float* eap;
    if (e < N_EDGES) { sn = src[e]; dn = dst[e]; eap = ea + (size_t)e * F_EDGE; }
    else             { sn = dn = e - N_EDGES;    eap = eamean; }
    float eav[F_EDGE];
#pragma unroll
    for (int f = 0; f < F_EDGE; ++f) eav[f] = eap[f];
    const int col0 = lane * 2;
    float acc = 0.0f;
#pragma unroll
    for (int j = 0; j < 2; ++j) {
        int col = col0 + j;
        float eev = 0.0f;
#pragma unroll
        for (int f = 0; f < F_EDGE; ++f) eev += eav[f] * We2[f * 64 + col];
        float m = xl[(size_t)sn * 64 + col] + xr[(size_t)dn * 64 + col] + eev;
        m = (m > 0.0f) ? m : NEG_SLOPE * m;
        acc += m * att2[col];
    }
    acc += __shfl_xor(acc, 1, 32);
    acc += __shfl_xor(acc, 2, 32);
    acc += __shfl_xor(acc, 4, 32);
    acc += __shfl_xor(acc, 8, 32);
    acc += __shfl_xor(acc, 16, 32);
    if (lane == 0) {
        s2[e] = acc;
        atomicMax(&smax2[dn], fenc(acc));
    }
}

__global__ void expden2_kernel(const int* __restrict__ dst, float* __restrict__ s2,
                               const unsigned* __restrict__ smax2, float* __restrict__ den2) {
    long long i = (long long)blockIdx.x * blockDim.x + threadIdx.x;
    if (i >= EPRIME) return;
    int e = (int)i;
    int dn = (e < N_EDGES) ? dst[e] : e - N_EDGES;
    float ez = __expf(s2[e] - fdec(smax2[dn]));
    s2[e] = ez;
    atomicAdd(&den2[dn], ez);
}

__global__ void scatter2_kernel(const float* __restrict__ xl,
                                const int* __restrict__ src, const int* __restrict__ dst,
                                const float* __restrict__ s2, const float* __restrict__ den2,
                                float* __restrict__ out2) {
    int e = blockIdx.x * (blockDim.x >> 5) + (threadIdx.x >> 5);
    if (e >= EPRIME) return;
    const int lane = threadIdx.x & 31;
    int sn, dn;
    if (e < N_EDGES) { sn = src[e]; dn = dst[e]; } else { sn = dn = e - N_EDGES; }
    float alpha = s2[e] / (den2[dn] + 1e-16f);
    const float* xs = xl + (size_t)sn * 64 + lane * 2;
    float* od = out2 + (size_t)dn * 64 + lane * 2;
    atomicAdd(&od[0], alpha * xs[0]);
    atomicAdd(&od[1], alpha * xs[1]);
}

// ---------------------------------------------------------------------
// Node classifier: logits = relu(x2@Wn1+bn1)@Wn2+bn2. Tiny; plain VALU.
__global__ void node_cls_kernel(const float* __restrict__ x2,
                                const float* __restrict__ Wn1, const float* __restrict__ bn1,
                                const float* __restrict__ Wn2, const float* __restrict__ bn2,
                                float* __restrict__ out) {
    int n = blockIdx.x * blockDim.x + threadIdx.x;
    if (n >= N_NODES) return;
    float xrow[64];
    const float4* xp = (const float4*)(x2 + (size_t)n * 64);
#pragma unroll
    for (int k = 0; k < 16; ++k) {
        float4 t = xp[k];
        xrow[4*k] = t.x; xrow[4*k+1] = t.y; xrow[4*k+2] = t.z; xrow[4*k+3] = t.w;
    }
    float logit = bn2[0];
    for (int j = 0; j < 32; ++j) {
        float h = bn1[j];
#pragma unroll
        for (int k = 0; k < 64; ++k) h += xrow[k] * Wn1[k * 32 + j];
        logit += fmaxf(h, 0.0f) * Wn2[j];
    }
    out[n] = logit;
}

// ---------------------------------------------------------------------
// Edge classifier: relu([x2[src],x2[dst],ea] @ Wec1 + bec1) @ Wec2 + bec2.
// Gathered-A WMMA GEMM; Wec1 (144x64) staged in LDS as float2 pairs (36 KB).
// One wave computes a 16(edge)x64 tile (4 accumulators), then fused
// ReLU + 64->1 dot + shfl reduction. K boundaries (64,128) are multiples
// of 4 so the A-source branch is wave-uniform per k-step.
__global__ void edge_cls_wmma(const float* __restrict__ x2, const float* __restrict__ ea,
                              const int* __restrict__ src, const int* __restrict__ dst,
                              const float* __restrict__ Wec1, const float* __restrict__ bec1,
                              const float* __restrict__ Wec2, const float* __restrict__ bec2,
                              float* __restrict__ elog) {
    __shared__ float2 Bs[72 * 64];             // (144/2) x 64 pairs = 36 KB
    for (int idx = threadIdx.x; idx < 72 * 64; idx += blockDim.x) {
        int kk = idx >> 6, c = idx & 63;
        float2 t;
        t.x = Wec1[(size_t)(2 * kk)     * 64 + c];
        t.y = Wec1[(size_t)(2 * kk + 1) * 64 + c];
        Bs[idx] = t;
    }
    __syncthreads();

    const int lane = threadIdx.x & 31;
    int tm = blockIdx.x * (blockDim.x >> 5) + (threadIdx.x >> 5);
    if (tm >= N_EDGES / 16) return;            // wave-uniform
    const int r  = lane & 15;
    const int kh = (lane >> 4) << 1;
    const int e  = tm * 16 + r;
    const float* xs  = x2 + (size_t)src[e] * 64;
    const float* xd  = x2 + (size_t)dst[e] * 64;
    const float* eae = ea + (size_t)e * F_EDGE;
    v8f acc[4] = {{}, {}, {}, {}};
    for (int k0 = 0; k0 < 144; k0 += 4) {
        const int k = k0 + kh;
        v2f a;
        if (k < 64)       { a.x = xs[k];        a.y = xs[k + 1];    }
        else if (k < 128) { a.x = xd[k - 64];   a.y = xd[k - 63];   }
        else              { a.x = eae[k - 128]; a.y = eae[k - 127]; }
        const float2* bp = Bs + ((k >> 1) << 6);
#pragma unroll
        for (int t = 0; t < 4; ++t) {
            float2 bt = bp[t * 16 + r];
            v2f b; b.x = bt.x; b.y = bt.y;
            acc[t] = __builtin_amdgcn_wmma_f32_16x16x4_f32(false, a, false, b,
                                                           (short)0, acc[t], false, false);
        }
    }
    float bv[4], wv[4];
#pragma unroll
    for (int t = 0; t < 4; ++t) { bv[t] = bec1[t * 16 + r]; wv[t] = Wec2[t * 16 + r]; }
    const float b2 = bec2[0];
#pragma unroll
    for (int v = 0; v < 8; ++v) {
        float p = 0.0f;
#pragma unroll
        for (int t = 0; t < 4; ++t) p += fmaxf(acc[t][v] + bv[t], 0.0f) * wv[t];
        p += __shfl_xor(p, 1, 32);
        p += __shfl_xor(p, 2, 32);
        p += __shfl_xor(p, 4, 32);
        p += __shfl_xor(p, 8, 32);
        if ((lane & 15) == 0)
            elog[tm * 16 + v + ((lane >> 4) << 3)] = p + b2;
    }
}

// ---------------------------------------------------------------------
extern "C" void kernel_launch(void* const* d_in, const int* in_sizes, int n_in,
                              void* d_out, int out_size, void* d_ws, size_t ws_size,
                              hipStream_t stream) {
    const float* x      = (const float*)d_in[0];
    const int*   eidx   = (const int*)d_in[1];
    const float* ea     = (const float*)d_in[2];
    const float* Wl1    = (const float*)d_in[3];
    const float* bl1    = (const float*)d_in[4];
    const float* Wr1    = (const float*)d_in[5];
    const float* br1    = (const float*)d_in[6];
    const float* We1    = (const float*)d_in[7];
    const float* att1   = (const float*)d_in[8];
    const float* bias1  = (const float*)d_in[9];
    const float* Wl2    = (const float*)d_in[10];
    const float* bl2    = (const float*)d_in[11];
    const float* Wr2    = (const float*)d_in[12];
    const float* br2    = (const float*)d_in[13];
    const float* We2    = (const float*)d_in[14];
    const float* att2   = (const float*)d_in[15];
    const float* bias2  = (const float*)d_in[16];
    const float* Wn1    = (const float*)d_in[17];
    const float* bn1    = (const float*)d_in[18];
    const float* Wn2    = (const float*)d_in[19];
    const float* bn2    = (const float*)d_in[20];
    const float* Wec1   = (const float*)d_in[21];
    const float* bec1   = (const float*)d_in[22];
    const float* Wec2   = (const float*)d_in[23];
    const float* bec2   = (const float*)d_in[24];

    const int* src = eidx;             // edge_index[0]
    const int* dst = eidx + N_EDGES;   // edge_index[1]

    float* node_logits = (float*)d_out;
    float* edge_logits = node_logits + N_NODES;
    float* x2out       = edge_logits + N_EDGES;   // [N,64] output #3

    // ---- workspace carve-up (xr1 reused as out1, xr2 as out2) ----
    float* w = (float*)d_ws;
    size_t o = 0;
    float*    xl1   = w + o;              o += (size_t)N_NODES * 256;
    float*    xr1   = w + o;              o += (size_t)N_NODES * 256;  // -> out1
    float*    x1    = w + o;              o += (size_t)N_NODES * 256;
    float*    xl2   = w + o;              o += (size_t)N_NODES * 64;
    float*    xr2   = w + o;              o += (size_t)N_NODES * 64;   // -> out2
    float*    s1    = w + o;              o += (size_t)EPRIME * HEADS;
    unsigned* smax1 = (unsigned*)(w + o); o += (size_t)N_NODES * HEADS;
    float*    den1  = w + o;              o += (size_t)N_NODES * HEADS;
    float*    s2    = w + o;              o += (size_t)EPRIME;
    unsigned* smax2 = (unsigned*)(w + o); o += (size_t)N_NODES;
    float*    den2  = w + o;              o += (size_t)N_NODES;
    float*    eamean= w + o;              o += 32;
    float*    out1  = xr1;
    float*    out2  = xr2;

    const int T = 256;                 // 8 waves per block (wave32)
    const int WPB = T / 32;

    // ---- edge_attr mean (for self-loop fill) ----
    zero_f32<<<1, 32, 0, stream>>>(eamean, 32);
    easum_kernel<<<64, T, 0, stream>>>(ea, eamean);
    eamean_div_kernel<<<1, 32, 0, stream>>>(eamean);

    // ---- layer 1 dense transforms (WMMA f32 16x16x4, LDS-staged weights) ----
    {
        int blks = CEILDIV(N_NODES, 128) * (256 / 64);   // row-groups x col-groups
        size_t ldsb = (size_t)(64 / 2) * 64 * sizeof(float2);   // K=64 -> 16 KB
        gemm_wmma_lds<<<blks, T, ldsb, stream>>>(x, Wl1, bl1, xl1, N_NODES, 64, 256, 0);
        gemm_wmma_lds<<<blks, T, ldsb, stream>>>(x, Wr1, br1, xr1, N_NODES, 64, 256, 0);
    }

    // ---- layer 1 attention ----
    zero_f32<<<CEILDIV(N_NODES * HEADS, T), T, 0, stream>>>((float*)smax1, (long long)N_NODES * HEADS);
    zero_f32<<<CEILDIV(N_NODES * HEADS, T), T, 0, stream>>>(den1, (long long)N_NODES * HEADS);
    {
        int blks = CEILDIV(EPRIME, WPB);
        scores1_kernel<<<blks, T, 0, stream>>>(xl1, xr1, src, dst, ea, eamean,
                                               We1, att1, s1, smax1);
        // xr1 dead after scores1 -> becomes out1
        zero_f32<<<CEILDIV((long long)N_NODES * 256, T), T, 0, stream>>>(out1, (long long)N_NODES * 256);
        expden1_kernel<<<CEILDIV((long long)EPRIME * HEADS, T), T, 0, stream>>>(dst, s1, smax1, den1);
        scatter1_kernel<<<blks, T, 0, stream>>>(xl1, src, dst, s1, den1, out1);
        addbias_elu_kernel<<<CEILDIV((long long)N_NODES * 256, T), T, 0, stream>>>(
            out1, bias1, x1, (long long)N_NODES * 256, 255);
    }

    // ---- layer 2 dense transforms ----
    {
        int blks = CEILDIV(N_NODES, 128) * (64 / 64);
        size_t ldsb = (size_t)(256 / 2) * 64 * sizeof(float2);  // K=256 -> 64 KB
        gemm_wmma_lds<<<blks, T, ldsb, stream>>>(x1, Wl2, bl2, xl2, N_NODES, 256, 64, 0);
        gemm_wmma_lds<<<blks, T, ldsb, stream>>>(x1, Wr2, br2, xr2, N_NODES, 256, 64, 0);
    }

    // ---- layer 2 attention ----
    zero_f32<<<CEILDIV(N_NODES, T), T, 0, stream>>>((float*)smax2, N_NODES);
    zero_f32<<<CEILDIV(N_NODES, T), T, 0, stream>>>(den2, N_NODES);
    {
        int blks = CEILDIV(EPRIME, WPB);
        scores2_kernel<<<blks, T, 0, stream>>>(xl2, xr2, src, dst, ea, eamean,
                                               We2, att2, s2, smax2);
        zero_f32<<<CEILDIV((long long)N_NODES * 64, T), T, 0, stream>>>(out2, (long long)N_NODES * 64);
        expden2_kernel<<<CEILDIV(EPRIME, T), T, 0, stream>>>(dst, s2, smax2, den2);
        scatter2_kernel<<<blks, T, 0, stream>>>(xl2, src, dst, s2, den2, out2);
        addbias_elu_kernel<<<CEILDIV((long long)N_NODES * 64, T), T, 0, stream>>>(
            out2, bias2, x2out, (long long)N_NODES * 64, 63);
    }

    // ---- classifiers ----
    node_cls_kernel<<<CEILDIV(N_NODES, T), T, 0, stream>>>(x2out, Wn1, bn1, Wn2, bn2, node_logits);
    {
        int blks = CEILDIV(N_EDGES / 16, WPB);
        edge_cls_wmma<<<blks, T, 0, stream>>>(x2out, ea, src, dst,
                                              Wec1, bec1, Wec2, bec2, edge_logits);
    }
}